// SubspaceMethod_53755810677125
// MI455X (gfx1250) — compile-verified
//
#include <hip/hip_runtime.h>
#include <hip/hip_bf16.h>

typedef float v2f __attribute__((ext_vector_type(2)));
typedef float v8f __attribute__((ext_vector_type(8)));

// One wave (32 lanes) per batch item.
// Phase 1: complex covariance Rx = x x^H / T via V_WMMA_F32_16X16X4_F32.
//          Re(Rx)*T = P = xre.xre^T + xim.xim^T  (2 WMMAs / k-block)
//          Im(Rx)*T = Q - Q^T with Q = xim.xre^T (1 WMMA / k-block; transpose is free in LDS)
//          P and Q are each split into two accumulators to break the D->C
//          dependency chain between back-to-back WMMAs.
// Phase 2: two-sided complex Jacobi eigensolve of the 16x16 Hermitian Rx in LDS.
// Phase 3: sort, subspace split, source estimation, per-batch regularizer term.
__global__ __launch_bounds__(32) void subspace_main(
    const float* __restrict__ xre, const float* __restrict__ xim,
    const int* __restrict__ nsrc, float* __restrict__ out,
    float* __restrict__ regws, int nb)
{
    const int b    = blockIdx.x;
    const int lane = threadIdx.x;
    const int r    = lane & 15;   // row (A) / col (B) index within tile
    const int h    = lane >> 4;   // half-wave select

    const float* prer = xre + ((size_t)b * 16 + r) * 256;
    const float* pimr = xim + ((size_t)b * 16 + r) * 256;

    // Split accumulators: P = P0+P1 (symmetric), Q = Q0+Q1 with Im*T = Q - Q^T
    v8f P0 = {0.f,0.f,0.f,0.f,0.f,0.f,0.f,0.f};
    v8f P1 = {0.f,0.f,0.f,0.f,0.f,0.f,0.f,0.f};
    v8f Q0 = {0.f,0.f,0.f,0.f,0.f,0.f,0.f,0.f};
    v8f Q1 = {0.f,0.f,0.f,0.f,0.f,0.f,0.f,0.f};

    // 16x16x4 f32 WMMA fragments: A lane(L<16) holds {k0,k1}, lane(L>=16) {k2,k3};
    // B mirrors it with N=lane. Loading identical per-lane row data for A and B
    // yields sum_t x[m][t]*y[n][t]. K-order permutation is legal (pure sum).
    #pragma unroll 4
    for (int t0 = 0; t0 < 256; t0 += 8) {
        float4 xr = *(const float4*)(prer + t0 + 4 * h);
        float4 xi = *(const float4*)(pimr + t0 + 4 * h);
        v2f arA = {xr.x, xr.y}, aiA = {xi.x, xi.y};
        v2f arB = {xr.z, xr.w}, aiB = {xi.z, xi.w};
        P0 = __builtin_amdgcn_wmma_f32_16x16x4_f32(false, arA, false, arA, (short)0, P0, false, false);
        P1 = __builtin_amdgcn_wmma_f32_16x16x4_f32(false, arB, false, arB, (short)0, P1, false, false);
        Q0 = __builtin_amdgcn_wmma_f32_16x16x4_f32(false, aiA, false, arA, (short)0, Q0, false, false);
        Q1 = __builtin_amdgcn_wmma_f32_16x16x4_f32(false, aiB, false, arB, (short)0, Q1, false, false);
        P0 = __builtin_amdgcn_wmma_f32_16x16x4_f32(false, aiA, false, aiA, (short)0, P0, false, false);
        P1 = __builtin_amdgcn_wmma_f32_16x16x4_f32(false, aiB, false, aiB, (short)0, P1, false, false);
    }
    v8f P = P0 + P1;
    v8f Q = Q0 + Q1;

    __shared__ float Are[16][17], Aim[16][17];
    __shared__ float Vre[16][17], Vim[16][17];
    __shared__ float ev[16];
    __shared__ int   ord[16];

    // C/D layout: VGPR j, lanes 0-15 -> M=j, N=lane ; lanes 16-31 -> M=j+8, N=lane-16.
    // First drop raw P into Are and raw Q into Aim, then build the exact
    // Hermitian matrix: Are = (P + P^T)/2 * invT, Aim = (Q - Q^T) * invT.
    const float invT = 1.0f / 256.0f;
    #pragma unroll
    for (int j = 0; j < 8; ++j) {
        int m = j + 8 * h;
        Are[m][r] = P[j];
        Aim[m][r] = Q[j];
        Vre[m][r] = (m == r) ? 1.0f : 0.0f;
        Vim[m][r] = 0.0f;
    }
    __syncthreads();
    float pm[8], pt[8], qm[8], qt[8];
    #pragma unroll
    for (int j = 0; j < 8; ++j) {
        int m = j + 8 * h;
        pm[j] = Are[m][r]; pt[j] = Are[r][m];
        qm[j] = Aim[m][r]; qt[j] = Aim[r][m];
    }
    __syncthreads();
    #pragma unroll
    for (int j = 0; j < 8; ++j) {
        int m = j + 8 * h;
        Are[m][r] = 0.5f * (pm[j] + pt[j]) * invT;   // symmetric real part
        Aim[m][r] = (qm[j] - qt[j]) * invT;          // antisymmetric imag part (diag == 0 exactly)
    }
    __syncthreads();

    // Two-sided complex Jacobi: J = diag(1, e^{-ia}) * Givens(c,s),  A <- J^H A J, V <- V J.
    for (int sweep = 0; sweep < 8; ++sweep) {
        for (int p = 0; p < 15; ++p) {
            for (int q = p + 1; q < 16; ++q) {
                float app = Are[p][p], aqq = Are[q][q];
                float zr  = Are[p][q], zi  = Aim[p][q];
                float mag2 = zr * zr + zi * zi;          // uniform across the wave
                if (mag2 > 1e-24f) {
                    float mag = sqrtf(mag2);
                    float er  =  zr / mag;               // e^{-ia} = (er, ei)
                    float ei  = -zi / mag;
                    float tau = (aqq - app) / (2.0f * mag);
                    float tt  = (tau >= 0.0f ? 1.0f : -1.0f) /
                                (fabsf(tau) + sqrtf(1.0f + tau * tau));
                    float c   = rsqrtf(1.0f + tt * tt);
                    float s   = tt * c;
                    // Column update: lanes 0-15 on A, lanes 16-31 on V (no overlap).
                    if (h == 0) {
                        float xpr = Are[r][p], xpi = Aim[r][p];
                        float xqr = Are[r][q], xqi = Aim[r][q];
                        float eqr = er * xqr - ei * xqi;   // e^{-ia} * A[r][q]
                        float eqi = er * xqi + ei * xqr;
                        Are[r][p] = c * xpr - s * eqr;
                        Aim[r][p] = c * xpi - s * eqi;
                        Are[r][q] = s * xpr + c * eqr;
                        Aim[r][q] = s * xpi + c * eqi;
                    } else {
                        float xpr = Vre[r][p], xpi = Vim[r][p];
                        float xqr = Vre[r][q], xqi = Vim[r][q];
                        float eqr = er * xqr - ei * xqi;
                        float eqi = er * xqi + ei * xqr;
                        Vre[r][p] = c * xpr - s * eqr;
                        Vim[r][p] = c * xpi - s * eqi;
                        Vre[r][q] = s * xpr + c * eqr;
                        Vim[r][q] = s * xpi + c * eqi;
                    }
                    __syncthreads();
                    // Row update: rows p,q with e^{+ia} = (er, -ei).
                    if (h == 0) {
                        float ypr = Are[p][r], ypi = Aim[p][r];
                        float yqr = Are[q][r], yqi = Aim[q][r];
                        float fr = er, fi = -ei;
                        float eqr = fr * yqr - fi * yqi;
                        float eqi = fr * yqi + fi * yqr;
                        Are[p][r] = c * ypr - s * eqr;
                        Aim[p][r] = c * ypi - s * eqi;
                        Are[q][r] = s * ypr + c * eqr;
                        Aim[q][r] = s * ypi + c * eqi;
                    }
                    __syncthreads();
                }
            }
        }
    }
    __syncthreads();

    // Sort eigenvalues descending (lane 0, tiny n=16 selection sort).
    if (lane == 0) {
        for (int i = 0; i < 16; ++i) { ev[i] = Are[i][i]; ord[i] = i; }
        for (int i = 0; i < 15; ++i) {
            int mx = i;
            for (int j = i + 1; j < 16; ++j)
                if (ev[ord[j]] > ev[ord[mx]]) mx = j;
            int t = ord[i]; ord[i] = ord[mx]; ord[mx] = t;
        }
    }
    __syncthreads();

    const int ns = nsrc[0];
    const float inv_lmax = 1.0f / ev[ord[0]];

    if (lane == 0) {
        int cnt = 0;
        for (int i = 0; i < 16; ++i)
            cnt += ((ev[ord[i]] * inv_lmax - 0.5f) > 0.0f) ? 1 : 0;
        out[(size_t)nb * 512 + b] = (float)cnt;           // source_estimation
        float a1 = ev[ord[ns - 1]] * inv_lmax - 0.5f;
        float a2 = ev[ord[ns]]     * inv_lmax - 0.5f;
        regws[b] = a1 * a2;                               // per-batch reg term
    }

    // Subspaces: complex64 interleaved (re, im). Signal (B,16,ns), noise (B,16,16-ns).
    if (h == 0) {
        size_t sbase = (size_t)(b * 16 + r) * ns * 2;
        for (int j = 0; j < ns; ++j) {
            int cq = ord[j];
            out[sbase + 2 * j]     = Vre[r][cq];
            out[sbase + 2 * j + 1] = Vim[r][cq];
        }
        int nn = 16 - ns;
        size_t nbase = (size_t)nb * 16 * ns * 2 + (size_t)(b * 16 + r) * nn * 2;
        for (int j = 0; j < nn; ++j) {
            int cq = ord[ns + j];
            out[nbase + 2 * j]     = Vre[r][cq];
            out[nbase + 2 * j + 1] = Vim[r][cq];
        }
    }
}

// Deterministic fixed-order reduction of the per-batch regularizer terms.
__global__ __launch_bounds__(256) void reg_reduce(
    const float* __restrict__ ws, float* __restrict__ out, int nb)
{
    __shared__ float sm[256];
    int t = threadIdx.x;
    float acc = 0.0f;
    for (int i = t; i < nb; i += 256) acc += ws[i];
    sm[t] = acc;
    __syncthreads();
    for (int sft = 128; sft > 0; sft >>= 1) {
        if (t < sft) sm[t] += sm[t + sft];
        __syncthreads();
    }
    if (t == 0) out[(size_t)nb * 513] = sm[0];
}

extern "C" void kernel_launch(void* const* d_in, const int* in_sizes, int n_in,
                              void* d_out, int out_size, void* d_ws, size_t ws_size,
                              hipStream_t stream)
{
    const float* xre  = (const float*)d_in[0];
    const float* xim  = (const float*)d_in[1];
    const int*   nsrc = (const int*)d_in[2];
    float* out = (float*)d_out;
    float* ws  = (float*)d_ws;   // nb floats of scratch for the reg terms

    const int nb = in_sizes[0] / (16 * 256);   // batch count (8192)

    hipLaunchKernelGGL(subspace_main, dim3(nb), dim3(32), 0, stream,
                       xre, xim, nsrc, out, ws, nb);
    hipLaunchKernelGGL(reg_reduce, dim3(1), dim3(256), 0, stream, ws, out, nb);
}